// GraphCNNLayer2_1_24919400252237
// MI455X (gfx1250) — compile-verified
//
#include <hip/hip_runtime.h>
#include <hip/hip_bf16.h>

typedef float v2f __attribute__((ext_vector_type(2)));
typedef float v8f __attribute__((ext_vector_type(8)));

#define B_   4
#define N_   1024
#define L_   8
#define F_   64
#define AO_  8
#define EPS_ 1e-5f

// ---------------------------------------------------------------------------
// 1) sq[b*N + n] = sum_f V[b,n,f]^2
// ---------------------------------------------------------------------------
__global__ void sq_kernel(const float* __restrict__ V, float* __restrict__ sq) {
    int idx = blockIdx.x * blockDim.x + threadIdx.x;        // 0 .. B*N-1
    const float4* row = (const float4*)(V + (size_t)idx * F_);
    float s = 0.0f;
#pragma unroll
    for (int i = 0; i < F_ / 4; ++i) {
        float4 v = row[i];
        s += v.x * v.x + v.y * v.y + v.z * v.z + v.w * v.w;
    }
    sq[idx] = s;
}

// ---------------------------------------------------------------------------
// 2) gaus[b,n,m] = exp(2*dot(V[b,n],V[b,m]) - sq_n - sq_m)
//    One 16x16 tile per wave, built from 16x V_WMMA_F32_16X16X4_F32 (K=4).
//    A-frag (16x4 f32): lane L -> (M = L%16, K = 2*(L/16)+v)  == float2 load
//    B-frag (4x16 f32): lane L -> (K = 2*(L/16)+v, N = L%16)  == same pattern
//    C/D   (16x16 f32): vgpr v, lane L -> (M = v + 8*(L/16), N = L%16)
// ---------------------------------------------------------------------------
__global__ void gaus_kernel(const float* __restrict__ V,
                            const float* __restrict__ sq,
                            float* __restrict__ gaus) {
    const int lane = threadIdx.x & 31;
    const int wave = threadIdx.x >> 5;
    const int tile = blockIdx.x * (blockDim.x >> 5) + wave; // 0 .. B*64*64-1
    const int b  = tile >> 12;                              // / 4096
    const int r  = tile & 4095;
    const int n0 = (r >> 6) << 4;
    const int m0 = (r & 63) << 4;
    const int half = lane >> 4;                             // 0 or 1
    const int lm   = lane & 15;

    const float* Vb = V + (size_t)b * N_ * F_;
    const float* An = Vb + (size_t)(n0 + lm) * F_;
    const float* Bm = Vb + (size_t)(m0 + lm) * F_;

    v8f acc = {};
#pragma unroll
    for (int k = 0; k < F_; k += 4) {
        v2f av = *(const v2f*)(An + k + 2 * half);
        v2f bv = *(const v2f*)(Bm + k + 2 * half);
        acc = __builtin_amdgcn_wmma_f32_16x16x4_f32(
            /*neg_a=*/false, av, /*neg_b=*/false, bv,
            /*c_mod=*/(short)0, acc, /*reuse_a=*/false, /*reuse_b=*/false);
    }

    const float sqn = sq[b * N_ + n0 + lm];   // lane lm holds sq[n0+lm]
    const float sqm = sq[b * N_ + m0 + lm];   // this lane's column norm
    float* grow = gaus + ((size_t)b * N_ + n0) * N_ + m0 + lm;
#pragma unroll
    for (int v = 0; v < 8; ++v) {
        float srow = __shfl(sqn, v + 8 * half, 32);         // sq[n0 + v + 8*half]
        float g = __expf(2.0f * acc[v] - srow - sqm);
        grow[(size_t)(v + 8 * half) * N_] = g;
    }
}

// ---------------------------------------------------------------------------
// 3) pass1: one block per (b,n). out = relu(A1 . W + bias) (unnormalized),
//    plus deterministic per-channel sum / sum^2 partials per block.
// ---------------------------------------------------------------------------
__global__ void pass1_kernel(const float* __restrict__ A,
                             const float* __restrict__ gaus,
                             const float* __restrict__ W,
                             const float* __restrict__ bias,
                             float* __restrict__ out,
                             float* __restrict__ partials) {
    __shared__ float sW[AO_ * (L_ + 1)];
    __shared__ float sB[AO_];
    __shared__ float wred[8][16];

    const int tid = threadIdx.x;
    if (tid < AO_ * (L_ + 1)) sW[tid] = W[tid];
    if (tid < AO_)            sB[tid] = bias[tid];
    __syncthreads();

    const int bn = blockIdx.x;                              // 0 .. B*N-1
    const float* Arow = A    + (size_t)bn * L_ * N_;
    const float* grow = gaus + (size_t)bn * N_;
    float*       orow = out  + (size_t)bn * AO_ * N_;

    float s1[AO_] = {}, s2[AO_] = {};
#pragma unroll
    for (int j = 0; j < N_ / 256; ++j) {
        const int m = tid + j * 256;
        float a[L_ + 1];
#pragma unroll
        for (int l = 0; l < L_; ++l) a[l] = Arow[(size_t)l * N_ + m];
        a[L_] = grow[m];
#pragma unroll
        for (int o = 0; o < AO_; ++o) {
            float t = sB[o];
#pragma unroll
            for (int l = 0; l <= L_; ++l) t = fmaf(a[l], sW[o * (L_ + 1) + l], t);
            t = fmaxf(t, 0.0f);
            orow[(size_t)o * N_ + m] = t;
            s1[o] += t;
            s2[o] += t * t;
        }
    }

    // deterministic wave reduction, then cross-wave via LDS
#pragma unroll
    for (int o = 0; o < AO_; ++o) {
#pragma unroll
        for (int off = 16; off >= 1; off >>= 1) {
            s1[o] += __shfl_down(s1[o], off, 32);
            s2[o] += __shfl_down(s2[o], off, 32);
        }
    }
    const int lane = tid & 31, w = tid >> 5;
    if (lane == 0) {
#pragma unroll
        for (int o = 0; o < AO_; ++o) {
            wred[w][o]       = s1[o];
            wred[w][AO_ + o] = s2[o];
        }
    }
    __syncthreads();
    if (tid < 16) {
        float s = 0.0f;
#pragma unroll
        for (int ww = 0; ww < 8; ++ww) s += wred[ww][tid];
        partials[(size_t)blockIdx.x * 16 + tid] = s;
    }
}

// ---------------------------------------------------------------------------
// 4) stats: reduce 4096 x 16 partials -> scale[8], shift[8]
// ---------------------------------------------------------------------------
__global__ void stats_kernel(const float* __restrict__ partials,
                             const float* __restrict__ gamma,
                             const float* __restrict__ beta,
                             float* __restrict__ stats) {
    __shared__ float red[16][16];
    __shared__ float tot[16];
    const int q = threadIdx.x & 15;                          // quantity
    const int c = threadIdx.x >> 4;                          // chunk (16 x 256)
    float s = 0.0f;
    for (int i = c * 256; i < (c + 1) * 256; ++i) s += partials[(size_t)i * 16 + q];
    red[c][q] = s;
    __syncthreads();
    if (threadIdx.x < 16) {
        float t = 0.0f;
#pragma unroll
        for (int cc = 0; cc < 16; ++cc) t += red[cc][threadIdx.x];
        tot[threadIdx.x] = t;
    }
    __syncthreads();
    if (threadIdx.x < AO_) {
        const float inv_cnt = 1.0f / ((float)B_ * (float)N_ * (float)N_);
        float mean  = tot[threadIdx.x] * inv_cnt;
        float var   = tot[AO_ + threadIdx.x] * inv_cnt - mean * mean;
        float scale = gamma[threadIdx.x] * rsqrtf(var + EPS_);
        stats[threadIdx.x]       = scale;
        stats[AO_ + threadIdx.x] = beta[threadIdx.x] - mean * scale;
    }
}

// ---------------------------------------------------------------------------
// 5) pass2: in-place out = out*scale[o] + shift[o]  (8 floats / thread)
// ---------------------------------------------------------------------------
__global__ void pass2_kernel(float* __restrict__ out,
                             const float* __restrict__ stats) {
    const size_t idx = (size_t)blockIdx.x * blockDim.x + threadIdx.x;
    const int o = (int)((idx >> 7) & 7);                    // (idx*8/1024) % 8
    const float scale = stats[o];
    const float shift = stats[AO_ + o];
    float4* p = (float4*)(out + idx * 8);
    float4 a = p[0], b = p[1];
    a.x = fmaf(a.x, scale, shift); a.y = fmaf(a.y, scale, shift);
    a.z = fmaf(a.z, scale, shift); a.w = fmaf(a.w, scale, shift);
    b.x = fmaf(b.x, scale, shift); b.y = fmaf(b.y, scale, shift);
    b.z = fmaf(b.z, scale, shift); b.w = fmaf(b.w, scale, shift);
    p[0] = a; p[1] = b;
}

// ---------------------------------------------------------------------------
extern "C" void kernel_launch(void* const* d_in, const int* in_sizes, int n_in,
                              void* d_out, int out_size, void* d_ws, size_t ws_size,
                              hipStream_t stream) {
    const float* A     = (const float*)d_in[0];
    const float* V     = (const float*)d_in[1];
    const float* W     = (const float*)d_in[2];
    const float* bias  = (const float*)d_in[3];
    const float* gamma = (const float*)d_in[4];
    const float* beta  = (const float*)d_in[5];
    float* out = (float*)d_out;

    char*  ws       = (char*)d_ws;
    float* gaus     = (float*)ws;                                    // B*N*N  (16.8 MB)
    float* sq       = (float*)(ws + (size_t)B_ * N_ * N_ * sizeof(float));
    float* partials = sq + (size_t)B_ * N_;                          // 4096*16 floats
    float* stats    = partials + (size_t)B_ * N_ * 16;               // 16 floats

    sq_kernel   <<<(B_ * N_) / 256,            256, 0, stream>>>(V, sq);
    gaus_kernel <<<(B_ * 64 * 64) / 8,         256, 0, stream>>>(V, sq, gaus);
    pass1_kernel<<<B_ * N_,                    256, 0, stream>>>(A, gaus, W, bias, out, partials);
    stats_kernel<<<1,                          256, 0, stream>>>(partials, gamma, beta, stats);
    pass2_kernel<<<(size_t)(B_*N_*AO_*N_)/(256*8), 256, 0, stream>>>(out, stats);
}